// IncidenceConvolution_48326972015216
// MI455X (gfx1250) — compile-verified
//
#include <hip/hip_runtime.h>
#include <hip/hip_bf16.h>

typedef __attribute__((ext_vector_type(2))) float v2f;
typedef __attribute__((ext_vector_type(8))) float v8f;

// ---------------------------------------------------------------------------
// Stage 1: collapse the affine MLP to out = A*x + C using WMMA f32 16x16x4.
// Rows of the A-matrix tile: M=0 -> w1-chain, M=1 -> b1-chain, rows 2..15 = 0.
// One wave (32 threads) => EXEC is all 1s as WMMA requires.
// ---------------------------------------------------------------------------
__global__ __launch_bounds__(32) void cnf_prep_kernel(
    const float* __restrict__ w1, const float* __restrict__ b1,
    const float* __restrict__ w2, const float* __restrict__ b2,
    const float* __restrict__ w3, const float* __restrict__ b3,
    const float* __restrict__ w4, const float* __restrict__ b4,
    float* __restrict__ ac /* ws: ac[0]=A, ac[1]=C */) {
  const int lane = threadIdx.x;        // 0..31
  const int m    = lane & 15;          // M (A-tile row) / N-within-tile (B)
  const int kb   = (lane >> 4) * 2;    // K sub-offset: lanes 0-15 -> K{0,1}, 16-31 -> K{2,3}

  __shared__ float h0[32];             // running w1-chain row
  __shared__ float h1[32];             // running b1-chain row

  // ---- layer A: [w1;b1](2x32, zero-padded to 16x32) x w2(32x32) ----
  v8f acc0 = {}; v8f acc1 = {};
  #pragma unroll
  for (int kk = 0; kk < 8; ++kk) {
    const int k0 = kk * 4 + kb;
    v2f a;
    a.x = (m == 0) ? w1[k0]     : (m == 1) ? b1[k0]     : 0.0f;
    a.y = (m == 0) ? w1[k0 + 1] : (m == 1) ? b1[k0 + 1] : 0.0f;
    v2f bA, bB;                                   // B tiles: N = m / N = 16+m
    bA.x = w2[(k0    ) * 32 +      m];
    bA.y = w2[(k0 + 1) * 32 +      m];
    bB.x = w2[(k0    ) * 32 + 16 + m];
    bB.y = w2[(k0 + 1) * 32 + 16 + m];
    acc0 = __builtin_amdgcn_wmma_f32_16x16x4_f32(false, a, false, bA, (short)0, acc0, false, false);
    acc1 = __builtin_amdgcn_wmma_f32_16x16x4_f32(false, a, false, bB, (short)0, acc1, false, false);
  }
  // C/D layout: VGPR r, lanes 0-15 => M=r, N=lane.  Row0 = chain(w1), Row1 = chain(b1)+b2.
  if (lane < 16) {
    h0[lane]      = acc0[0];
    h0[lane + 16] = acc1[0];
    h1[lane]      = acc0[1] + b2[lane];
    h1[lane + 16] = acc1[1] + b2[lane + 16];
  }
  __syncthreads();

  // ---- layer B: [h0;h1](2x32) x w3(32x32) ----
  v8f d0 = {}; v8f d1 = {};
  #pragma unroll
  for (int kk = 0; kk < 8; ++kk) {
    const int k0 = kk * 4 + kb;
    v2f a;
    a.x = (m == 0) ? h0[k0]     : (m == 1) ? h1[k0]     : 0.0f;
    a.y = (m == 0) ? h0[k0 + 1] : (m == 1) ? h1[k0 + 1] : 0.0f;
    v2f bA, bB;
    bA.x = w3[(k0    ) * 32 +      m];
    bA.y = w3[(k0 + 1) * 32 +      m];
    bB.x = w3[(k0    ) * 32 + 16 + m];
    bB.y = w3[(k0 + 1) * 32 + 16 + m];
    d0 = __builtin_amdgcn_wmma_f32_16x16x4_f32(false, a, false, bA, (short)0, d0, false, false);
    d1 = __builtin_amdgcn_wmma_f32_16x16x4_f32(false, a, false, bB, (short)0, d1, false, false);
  }
  __syncthreads();
  if (lane < 16) {
    h0[lane]      = d0[0];
    h0[lane + 16] = d1[0];
    h1[lane]      = d0[1] + b3[lane];
    h1[lane + 16] = d1[1] + b3[lane + 16];
  }
  __syncthreads();

  // ---- final dot with w4 (wave32 shuffle reduction) ----
  float pa = h0[lane] * w4[lane];
  float pc = h1[lane] * w4[lane];
  #pragma unroll
  for (int off = 16; off > 0; off >>= 1) {
    pa += __shfl_down(pa, off);
    pc += __shfl_down(pc, off);
  }
  if (lane == 0) {
    ac[0] = pa;              // A
    ac[1] = pc + b4[0];      // C
  }
}

// ---------------------------------------------------------------------------
// Stage 2: HBM-bound pass. One block per [b,f] row; out = A*Q/S^2 + C with
// S = sum(x), Q = sum(x^2).  8 KB contiguous per block via float4 loads.
// ---------------------------------------------------------------------------
__global__ __launch_bounds__(256) void cnf_reduce_kernel(
    const float* __restrict__ x, const float* __restrict__ ac,
    float* __restrict__ out, int Lq /* L/4 */) {
  const int row = blockIdx.x;
  const float4* __restrict__ p4 =
      (const float4*)(x + (size_t)row * (size_t)Lq * 4u);
  const int t = threadIdx.x;

  float s = 0.0f, q = 0.0f;
  for (int i = t; i < Lq; i += 256) {
    float4 v = p4[i];
    s += (v.x + v.y) + (v.z + v.w);
    q += v.x * v.x + v.y * v.y + v.z * v.z + v.w * v.w;
  }

  // wave32 reduce, then LDS across the 8 waves
  #pragma unroll
  for (int off = 16; off > 0; off >>= 1) {
    s += __shfl_down(s, off);
    q += __shfl_down(q, off);
  }
  __shared__ float ss[8], qs[8];
  const int wid = t >> 5;
  if ((t & 31) == 0) { ss[wid] = s; qs[wid] = q; }
  __syncthreads();
  if (t < 8) {
    s = ss[t]; q = qs[t];
    #pragma unroll
    for (int off = 4; off > 0; off >>= 1) {
      s += __shfl_down(s, off);
      q += __shfl_down(q, off);
    }
    if (t == 0) {
      const float A = ac[0], C = ac[1];
      out[row] = A * q / (s * s) + C;
    }
  }
}

extern "C" void kernel_launch(void* const* d_in, const int* in_sizes, int n_in,
                              void* d_out, int out_size, void* d_ws, size_t ws_size,
                              hipStream_t stream) {
  const float* inc = (const float*)d_in[0];
  const float* w1  = (const float*)d_in[1];
  const float* b1  = (const float*)d_in[2];
  const float* w2  = (const float*)d_in[3];
  const float* b2  = (const float*)d_in[4];
  const float* w3  = (const float*)d_in[5];
  const float* b3  = (const float*)d_in[6];
  const float* w4  = (const float*)d_in[7];
  const float* b4  = (const float*)d_in[8];
  float* out = (float*)d_out;
  float* ac  = (float*)d_ws;                 // 2 floats of scratch

  const int rows = out_size;                 // B*F = 2048
  const int L    = in_sizes[0] / rows;       // 2048
  const int Lq   = L >> 2;                   // L % 4 == 0 in this problem

  cnf_prep_kernel<<<1, 32, 0, stream>>>(w1, b1, w2, b2, w3, b3, w4, b4, ac);
  cnf_reduce_kernel<<<rows, 256, 0, stream>>>(inc, ac, out, Lq);
}